// QueryAndGroup_16346645529142
// MI455X (gfx1250) — compile-verified
//
#include <hip/hip_runtime.h>
#include <stdint.h>

typedef __attribute__((ext_vector_type(2))) float v2f;
typedef __attribute__((ext_vector_type(8))) float v8f;

#define B_ 8
#define N_ 8192
#define M_ 2048
#define C_ 64
#define S_ 32
#define CH_ (3 + C_)
#define R2_ 1.0f

// ---------------------------------------------------------------------------
// Phase 1: ball query via V_WMMA_F32_16X16X4_F32 with double-buffered
// GLOBAL_LOAD_ASYNC_TO_LDS_B32 staging of the 16-point tiles (ASYNCcnt path).
// A row r  = [qx, qy, qz, 1] ; B col c = [-2px, -2py, -2pz, |p|^2]
// D[r][c]  = |p|^2 - 2 q.p   ; hit iff D < r^2 - |q|^2 (per-row threshold).
// D layout: VGPR j, lanes 0-15 -> row j; lanes 16-31 -> row j+8; col = lane%16.
// ---------------------------------------------------------------------------
__global__ __launch_bounds__(128) void ball_query_wmma(
    const float* __restrict__ xyz,      // [B][N][3]
    const float* __restrict__ new_xyz,  // [B][M][3]
    int* __restrict__ idx)              // [B][M][S]
{
  __shared__ float stage[4][2][64];     // per-wave double buffer (2 KB total)

  const int lane = threadIdx.x & 31;
  const int wave = threadIdx.x >> 5;
  const int tile = blockIdx.x * 4 + wave;          // b * (M/16) + mtile
  const int b  = tile / (M_ / 16);
  const int m0 = (tile % (M_ / 16)) * 16;

  const int r  = lane & 15;     // row (query) / col (point)
  const int hi = lane >> 4;

  // Query point for row r (both half-waves load the same 16 queries).
  const float* q = new_xyz + ((size_t)b * M_ + (m0 + r)) * 3;
  const float qx = q[0], qy = q[1], qz = q[2];
  const float q2 = qx * qx + qy * qy + qz * qz;

  // A fragment (16x4 f32): lanes 0-15 hold K=0,1; lanes 16-31 hold K=2,3.
  v2f afrag;
  afrag.x = hi ? qz   : qx;
  afrag.y = hi ? 1.0f : qy;

  // thrj[j] = r^2 - |q_{j+8*hi}|^2 for this lane's D half (ds_bpermute bcast).
  const float thr_lane = R2_ - q2;
  float thrj[8];
#pragma unroll
  for (int j = 0; j < 8; ++j)
    thrj[j] = __shfl(thr_lane, j + 8 * hi, 32);

  int cnt[16], first[16];
#pragma unroll
  for (int j = 0; j < 16; ++j) { cnt[j] = 0; first[j] = 0; }

  const float* pbase = xyz + (size_t)b * N_ * 3;
  const unsigned ltmask = (1u << lane) - 1u;
  const unsigned halfm  = hi ? 0xFFFF0000u : 0x0000FFFFu;
  int* const idxhalf = idx + ((size_t)b * M_ + m0 + 8 * hi) * S_;

  // Async staging: 48 dwords per tile; lane covers dword `lane` and 32+(lane&15)
  // (hi half duplicates the tail -> same-address LDS writes, benign).
  const int dw0 = lane;
  const int dw1 = 32 + r;

  auto issue_async = [&](int n0, int buf) {
    const float* g = pbase + (size_t)n0 * 3;
    unsigned l0 = (unsigned)(uintptr_t)&stage[wave][buf][dw0];
    unsigned l1 = (unsigned)(uintptr_t)&stage[wave][buf][dw1];
    unsigned long long g0 = (unsigned long long)(uintptr_t)(g + dw0);
    unsigned long long g1 = (unsigned long long)(uintptr_t)(g + dw1);
    asm volatile("global_load_async_to_lds_b32 %0, %1, off"
                 :: "v"(l0), "v"(g0) : "memory");
    asm volatile("global_load_async_to_lds_b32 %0, %1, off"
                 :: "v"(l1), "v"(g1) : "memory");
  };

  issue_async(0, 0);

  const v8f dzero = {0.f, 0.f, 0.f, 0.f, 0.f, 0.f, 0.f, 0.f};

  for (int it = 0; it < N_ / 16; ++it) {
    const int n0 = it * 16;
    int nnext = n0 + 16;
    if (nnext > N_ - 16) nnext = N_ - 16;      // clamped duplicate at the end
    issue_async(nnext, (it + 1) & 1);
    // Wait for the current tile (2 outstanding = the prefetch just issued).
    asm volatile("s_wait_asynccnt 0x2" ::: "memory");

    const float* sp = &stage[wave][it & 1][0];
    const float px = sp[r * 3 + 0];
    const float py = sp[r * 3 + 1];
    const float pz = sp[r * 3 + 2];
    const float p2 = px * px + py * py + pz * pz;
    v2f bfrag;
    bfrag.x = hi ? (-2.0f * pz) : (-2.0f * px);
    bfrag.y = hi ? p2           : (-2.0f * py);

    v8f d = __builtin_amdgcn_wmma_f32_16x16x4_f32(
        false, afrag, false, bfrag, (short)0, dzero, false, false);

#pragma unroll
    for (int j = 0; j < 8; ++j) {
      const bool pred = d[j] < thrj[j];
      const unsigned bal = (unsigned)__ballot(pred);
      const int slot = __popc(bal & halfm & ltmask);
      const int cbase = hi ? cnt[j + 8] : cnt[j];
      const int s = cbase + slot;
      if (pred & (s < S_))                      // single exec-mask branch
        idxhalf[j * S_ + s] = n0 + r;

      const unsigned lo16 = bal & 0xFFFFu;
      const unsigned hi16 = bal >> 16;
      first[j]     = (cnt[j]     == 0 && lo16)
                         ? (n0 + (int)__builtin_ctz(lo16 | 0x10000u)) : first[j];
      first[j + 8] = (cnt[j + 8] == 0 && hi16)
                         ? (n0 + (int)__builtin_ctz(hi16 | 0x10000u)) : first[j + 8];
      cnt[j]     += __popc(lo16);
      cnt[j + 8] += __popc(hi16);
    }

    bool done = true;
#pragma unroll
    for (int j = 0; j < 16; ++j) done = done && (cnt[j] >= S_);
    if (done) break;                            // uniform early exit
  }

  // Pad rows with < 32 hits using the first hit (reference semantics).
#pragma unroll
  for (int j = 0; j < 8; ++j) {
    const int c = hi ? cnt[j + 8] : cnt[j];
    const int f = hi ? first[j + 8] : first[j];
    for (int s = c + r; s < S_; s += 16) idxhalf[j * S_ + s] = f;
  }
}

// ---------------------------------------------------------------------------
// Phase 2: gather, 4 output elements per thread (b128 stores, int4 idx loads).
// out[b][ch][m][s]; ch<3 -> centered xyz, else features. Writes coalesced;
// scattered reads hit the 192MB L2 (features: 16.8MB resident).
// ---------------------------------------------------------------------------
__global__ __launch_bounds__(256) void group_gather(
    const float* __restrict__ xyz,      // [B][N][3]
    const float* __restrict__ new_xyz,  // [B][M][3]
    const float* __restrict__ feat,     // [B][C][N]
    const int* __restrict__ idx,        // [B][M][S]
    float* __restrict__ out)            // [B][CH][M][S]
{
  const size_t t = (size_t)blockIdx.x * blockDim.x + threadIdx.x;
  const size_t total4 = (size_t)B_ * CH_ * M_ * (S_ / 4);
  if (t >= total4) return;

  const int sg = (int)(t & 7) * 4;               // s0 in {0,4,...,28}
  size_t rem = t >> 3;
  const int m = (int)(rem % M_); rem /= M_;
  const int ch = (int)(rem % CH_);
  const int b = (int)(rem / CH_);

  const int4 nn = *(const int4*)&idx[((size_t)b * M_ + m) * S_ + sg];
  float4 v;
  if (ch < 3) {
    const float c = new_xyz[((size_t)b * M_ + m) * 3 + ch];
    const float* xb = xyz + (size_t)b * N_ * 3 + ch;
    v.x = xb[nn.x * 3] - c;
    v.y = xb[nn.y * 3] - c;
    v.z = xb[nn.z * 3] - c;
    v.w = xb[nn.w * 3] - c;
  } else {
    const float* f = feat + ((size_t)b * C_ + (ch - 3)) * N_;
    v.x = f[nn.x];
    v.y = f[nn.y];
    v.z = f[nn.z];
    v.w = f[nn.w];
  }
  *(float4*)&out[t * 4] = v;
}

extern "C" void kernel_launch(void* const* d_in, const int* in_sizes, int n_in,
                              void* d_out, int out_size, void* d_ws, size_t ws_size,
                              hipStream_t stream) {
  (void)in_sizes; (void)n_in; (void)out_size; (void)ws_size;
  const float* xyz     = (const float*)d_in[0];
  const float* new_xyz = (const float*)d_in[1];
  const float* feat    = (const float*)d_in[2];
  float* out = (float*)d_out;
  int* idx   = (int*)d_ws;   // B*M*S ints = 2 MB

  ball_query_wmma<<<dim3((B_ * (M_ / 16)) / 4), dim3(128), 0, stream>>>(
      xyz, new_xyz, idx);

  const size_t total4 = (size_t)B_ * CH_ * M_ * (S_ / 4);
  const int blocks = (int)((total4 + 255) / 256);
  group_gather<<<dim3(blocks), dim3(256), 0, stream>>>(
      xyz, new_xyz, feat, idx, out);
}